// MedSegNet_2199023256213
// MI455X (gfx1250) — compile-verified
//
#include <hip/hip_runtime.h>
#include <cstdint>

// Problem shape (fixed by the reference): x [8, 64, 128, 128] f32 -> out [8, 256, 128, 128] f32
#define BATCH 8
#define CHN   64
#define Hdim  128
#define Wdim  128
#define ROWS  8                       // output rows per block
#define TILE_R (ROWS + 2)             // 10 rows incl. halo
#define LDSW   136                    // padded LDS row stride (floats): 544 B, 16-B aligned rows
#define INT0   4                      // LDS col of gw=0  (left halo gw=-1 at j=3, right halo at j=132)
#define NTHREADS 256

// ---- gfx1250 async global->LDS staging (ASYNCcnt path), guarded so we always compile ----
#if defined(__gfx1250__) && __has_builtin(__builtin_amdgcn_global_load_async_to_lds_b32)
#define USE_ASYNC_LDS 1
typedef __attribute__((address_space(1))) int gbl_int;   // global int*
typedef __attribute__((address_space(3))) int lds_int;   // LDS int*
#else
#define USE_ASYNC_LDS 0
#endif
#if USE_ASYNC_LDS && __has_builtin(__builtin_amdgcn_global_load_async_to_lds_b128)
#define USE_ASYNC_B128 1
typedef int v4i_t __attribute__((__vector_size__(16)));  // 4 x i32, as the builtin expects
typedef __attribute__((address_space(1))) v4i_t gbl_v4i; // global v4i*
typedef __attribute__((address_space(3))) v4i_t lds_v4i; // LDS v4i*
#else
#define USE_ASYNC_B128 0
#endif

__device__ __forceinline__ float martingale(float f) {
  // exp(min(log(max(f,1e-5)) - 0.5, 15)) clipped to [1e-4, 1e4] with THETA=1
  // == clamp(max(f,1e-5) * e^{-0.5}, 1e-4, 1e4)  (exact; always finite)
  float m = fmaxf(f, 1e-5f) * 0.60653066f;
  return fminf(fmaxf(m, 1e-4f), 1e4f);
}

__global__ __launch_bounds__(NTHREADS)
void medseg_tex_kernel(const float* __restrict__ x, float* __restrict__ out)
{
  __shared__ __align__(16) float tile[TILE_R * LDSW];

  const int tid  = threadIdx.x;
  const int bc   = blockIdx.x >> 4;        // b*C + c   (H/ROWS == 16 strips)
  const int rblk = blockIdx.x & 15;
  const int r0   = rblk * ROWS;
  const float* src = x + (long)bc * (Hdim * Wdim);

  // Warm the strip's input lines (global_prefetch_b8) ahead of the async staging.
  if (tid < TILE_R * 4) {
    int i  = tid >> 2;
    int gh = r0 - 1 + i;
    if (gh >= 0 && gh < Hdim)
      __builtin_prefetch(src + gh * Wdim + (tid & 3) * 32, 0, 2);
  }

  // ---- Zero the statically-OOB cells (disjoint from async-written cells) ----
  // Halo columns gw=-1 (j=3) and gw=128 (j=132): always zero.
  if (tid < TILE_R * 2) {
    int i = tid >> 1;
    int j = (tid & 1) ? (INT0 + Wdim) : (INT0 - 1);
    tile[i * LDSW + j] = 0.0f;
  }
  // At most one OOB row per block (top strip: tile row 0; bottom strip: tile row 9).
  {
    int oob = (r0 == 0) ? 0 : ((r0 + ROWS == Hdim) ? (TILE_R - 1) : -1);
    if (oob >= 0 && tid < Wdim / 4) {
      float4 z = make_float4(0.f, 0.f, 0.f, 0.f);
      *(float4*)&tile[oob * LDSW + INT0 + tid * 4] = z;
    }
  }

  // ---- Async-stage interior: 10 rows x 32 x b128 (16 B per lane) ----
  #pragma unroll
  for (int t = tid; t < TILE_R * (Wdim / 4); t += NTHREADS) {   // 320 lane-tasks
    const int i    = t >> 5;
    const int lane = t & 31;
    const int gh   = r0 - 1 + i;
    if (gh >= 0 && gh < Hdim) {
      const float* gp = src + gh * Wdim + lane * 4;
      float*       lp = &tile[i * LDSW + INT0 + lane * 4];
#if USE_ASYNC_B128
      __builtin_amdgcn_global_load_async_to_lds_b128(
          (gbl_v4i*)(uintptr_t)gp, (lds_v4i*)(uintptr_t)lp, 0, 0);
#elif USE_ASYNC_LDS
      __builtin_amdgcn_global_load_async_to_lds_b32(
          (gbl_int*)(uintptr_t)(gp + 0), (lds_int*)(uintptr_t)(lp + 0), 0, 0);
      __builtin_amdgcn_global_load_async_to_lds_b32(
          (gbl_int*)(uintptr_t)(gp + 1), (lds_int*)(uintptr_t)(lp + 1), 0, 0);
      __builtin_amdgcn_global_load_async_to_lds_b32(
          (gbl_int*)(uintptr_t)(gp + 2), (lds_int*)(uintptr_t)(lp + 2), 0, 0);
      __builtin_amdgcn_global_load_async_to_lds_b32(
          (gbl_int*)(uintptr_t)(gp + 3), (lds_int*)(uintptr_t)(lp + 3), 0, 0);
#else
      *(float4*)lp = *(const float4*)gp;
#endif
    }
  }
#if USE_ASYNC_LDS
  asm volatile("s_wait_asynccnt 0" ::: "memory");
#endif
  __syncthreads();

  // ---- Compute: each thread owns one column, 4 consecutive rows ----
  const int col  = tid & (Wdim - 1);
  const int rsel = tid >> 7;               // 0 or 1
  const int base = rsel * (ROWS / 2);      // first local output row (0 or 4)
  float* outp = out + (long)(4 * bc) * (Hdim * Wdim);

  // Load the 6 tile rows this thread's 4 windows touch: 18 ds loads total.
  float rv[6][3];
  #pragma unroll
  for (int i = 0; i < 6; ++i)
    #pragma unroll
    for (int j = 0; j < 3; ++j)
      rv[i][j] = tile[(base + i) * LDSW + (col + INT0 - 1) + j];

  #pragma unroll
  for (int q = 0; q < ROWS / 2; ++q) {
    const int gh = r0 + base + q;

    float p[9];
    #pragma unroll
    for (int di = 0; di < 3; ++di)
      #pragma unroll
      for (int dj = 0; dj < 3; ++dj)
        p[di * 3 + dj] = rv[q + di][dj];

    float s1 = 0.f, s2 = 0.f;
    #pragma unroll
    for (int k = 0; k < 9; ++k) { s1 += p[k]; s2 += p[k] * p[k]; }
    const float mean = s1 * (1.0f / 9.0f);

    float sc2 = 0.f, sab = 0.f, se = 0.f;
    #pragma unroll
    for (int k = 0; k < 9; ++k) {
      float c = p[k] - mean;
      sc2 += c * c;
      sab += fabsf(c);
      float cl = fmaxf(p[k], 1e-6f);
      se  += cl * __logf(cl);              // v_log_f32 (the only transcendentals left)
    }

    const float var = sc2 * (1.0f / 8.0f);                 // unbiased (ddof=1)
    const float sd  = fmaxf(sqrtf(var), 1e-3f);
    const float contrast = sc2 / (9.0f * sd * sd);         // mean((centered/std)^2)
    const float energy   = s2 * (1.0f / 9.0f);
    const float entropy  = -se * (1.0f / 9.0f);
    const float homog    = 1.0f / (1.0f + sab * (1.0f / 9.0f) + 1e-6f);

    const long pix = (long)gh * Wdim + col;
    outp[0L * (Hdim * Wdim) + pix] = martingale(contrast);
    outp[1L * (Hdim * Wdim) + pix] = martingale(energy);
    outp[2L * (Hdim * Wdim) + pix] = martingale(entropy);
    outp[3L * (Hdim * Wdim) + pix] = martingale(homog);
  }
}

extern "C" void kernel_launch(void* const* d_in, const int* in_sizes, int n_in,
                              void* d_out, int out_size, void* d_ws, size_t ws_size,
                              hipStream_t stream) {
  (void)in_sizes; (void)n_in; (void)d_ws; (void)ws_size; (void)out_size;
  const float* x = (const float*)d_in[0];
  float* out = (float*)d_out;
  const int nblocks = BATCH * CHN * (Hdim / ROWS);   // 8192 blocks, 8 waves each
  medseg_tex_kernel<<<dim3(nblocks), dim3(NTHREADS), 0, stream>>>(x, out);
}